// ConvolutionBlock_54400055771686
// MI455X (gfx1250) — compile-verified
//
#include <hip/hip_runtime.h>
#include <stdint.h>

#define B_   8
#define V_   8192
#define VT   32            // vertices per block (tile)
#define NTH  256           // 8 wave32s per block

typedef uint32_t v4u  __attribute__((ext_vector_type(4)));
typedef int32_t  v8i_ __attribute__((ext_vector_type(8)));
typedef int32_t  v4i_ __attribute__((ext_vector_type(4)));

__global__ __launch_bounds__(NTH)
void bilinear_concat_kernel(const float* __restrict__ coords,
                            const float* __restrict__ fm3,
                            const float* __restrict__ fm4,
                            const float* __restrict__ fm5,
                            float* __restrict__ out)
{
    __shared__ float  s_coord[VT * 2];   // (x,y) per vertex in tile
    __shared__ int4   s_o[3][VT];        // 4 plane offsets per scale per vertex
    __shared__ float4 s_w[3][VT];        // 4 bilinear weights per scale per vertex

    const int tile = blockIdx.x;                 // 0 .. B*V/VT-1
    const int b    = tile / (V_ / VT);
    const int v0   = (tile % (V_ / VT)) * VT;
    const int tid  = threadIdx.x;

    // ---- Stage the 64 coordinate floats for this tile into LDS via the
    // ---- Tensor Data Mover (async DMA, tracked with TENSORcnt). Issued by
    // ---- wave 0 only (TDM ignores EXEC; other waves branch around).
    if (tid < 32) {
        const uint64_t gaddr = (uint64_t)(uintptr_t)(coords + ((size_t)b * V_ + v0) * 2);
        const uint32_t ldsa  = (uint32_t)(uintptr_t)(void*)&s_coord[0]; // low 32b of flat LDS addr = LDS offset

        v4u g0;
        g0[0] = 1u;                                                     // count=1, user descriptor
        g0[1] = ldsa;                                                   // lds_addr
        g0[2] = (uint32_t)(gaddr & 0xffffffffull);                      // global_addr[31:0]
        g0[3] = (uint32_t)((gaddr >> 32) & 0x01ffffffull) | (2u << 30); // global_addr[56:32] | type=2

        v8i_ g1;
        g1[0] = (int)(2u << 16);                   // data_size = 4 bytes; no multicast/pad/iterate
        g1[1] = (int)((uint32_t)(VT * 2) << 16);   // tensor_dim0 = 64 (lo16 at bits 63:48)
        g1[2] = (int)(1u << 16);                   // tensor_dim0 hi = 0 | tensor_dim1 = 1 (lo16)
        g1[3] = (int)((uint32_t)(VT * 2) << 16);   // tensor_dim1 hi = 0 | tile_dim0 = 64
        g1[4] = 1;                                 // tile_dim1 = 1, tile_dim2 = 0 (unused)
        g1[5] = VT * 2;                            // tensor_dim0_stride = 64 (lo32)
        g1[6] = 0;                                 // stride hi | tensor_dim1_stride lo
        g1[7] = 0;

        v4i_ g2 = {0, 0, 0, 0};                    // >2D dims unused
        v4i_ g3 = {0, 0, 0, 0};
        v8i_ g4 = {0, 0, 0, 0, 0, 0, 0, 0};        // trailing group (clang-23 6-arg form)

        __builtin_amdgcn_tensor_load_to_lds(g0, g1, g2, g3, g4, 0);
        __builtin_amdgcn_s_wait_tensorcnt((short)0);
    }
    __syncthreads();

    // ---- Per-(scale, vertex) corner offsets + weights (floor/ceil scheme,
    // ---- bit-matching the reference incl. zero weights at integer coords).
    if (tid < 3 * VT) {
        const int s = tid / VT;
        const int v = tid % VT;
        const float inv = (s == 0) ? 0.125f : (s == 1) ? 0.0625f : 0.03125f;
        const int   W   = (s == 0) ? 56     : (s == 1) ? 28      : 14;

        const float x = s_coord[2 * v + 0] * inv;
        const float y = s_coord[2 * v + 1] * inv;
        const float x1 = floorf(x), x2 = ceilf(x);
        const float y1 = floorf(y), y2 = ceilf(y);
        const int xi1 = (int)x1, xi2 = (int)x2;
        const int yi1 = (int)y1, yi2 = (int)y2;

        s_o[s][v] = make_int4(yi1 * W + xi1,   // pairs with w11=(x2-x)(y2-y)
                              yi2 * W + xi1,   // w12=(x2-x)(y-y1)
                              yi1 * W + xi2,   // w21=(x-x1)(y2-y)
                              yi2 * W + xi2);  // w22=(x-x1)(y-y1)
        const float ax = x2 - x, bx = x - x1;
        const float ay = y2 - y, by = y - y1;
        s_w[s][v] = make_float4(ax * ay, ax * by, bx * ay, bx * by);
    }
    __syncthreads();

    // ---- Each thread owns 5 fixed channel planes (concat layout 0/256/512/768/1024)
    // ---- and sweeps the 32 vertices: planes stay hot in L0/L2 across vertices.
    const float* __restrict__ p3  = fm3 + ((size_t)b * 256 + tid)       * (size_t)(56 * 56);
    const float* __restrict__ p4a = fm4 + ((size_t)b * 512 + tid)       * (size_t)(28 * 28);
    const float* __restrict__ p4b = fm4 + ((size_t)b * 512 + tid + 256) * (size_t)(28 * 28);
    const float* __restrict__ p5a = fm5 + ((size_t)b * 512 + tid)       * (size_t)(14 * 14);
    const float* __restrict__ p5b = fm5 + ((size_t)b * 512 + tid + 256) * (size_t)(14 * 14);
    float* __restrict__ orow = out + ((size_t)b * V_ + v0) * 1280 + tid;

#pragma unroll 2
    for (int v = 0; v < VT; ++v) {
        const int4   o3 = s_o[0][v]; const float4 w3 = s_w[0][v];
        const int4   o4 = s_o[1][v]; const float4 w4 = s_w[1][v];
        const int4   o5 = s_o[2][v]; const float4 w5 = s_w[2][v];

        const float r3  = p3 [o3.x] * w3.x + p3 [o3.y] * w3.y + p3 [o3.z] * w3.z + p3 [o3.w] * w3.w;
        const float r4a = p4a[o4.x] * w4.x + p4a[o4.y] * w4.y + p4a[o4.z] * w4.z + p4a[o4.w] * w4.w;
        const float r4b = p4b[o4.x] * w4.x + p4b[o4.y] * w4.y + p4b[o4.z] * w4.z + p4b[o4.w] * w4.w;
        const float r5a = p5a[o5.x] * w5.x + p5a[o5.y] * w5.y + p5a[o5.z] * w5.z + p5a[o5.w] * w5.w;
        const float r5b = p5b[o5.x] * w5.x + p5b[o5.y] * w5.y + p5b[o5.z] * w5.z + p5b[o5.w] * w5.w;

        // Non-temporal stores: 335 MB output stream must not evict the
        // L2-resident (41.7 MB) feature maps on the 192 MB L2.
        float* orow_v = orow + (size_t)v * 1280;
        __builtin_nontemporal_store(r3,  orow_v);
        __builtin_nontemporal_store(r4a, orow_v + 256);
        __builtin_nontemporal_store(r4b, orow_v + 512);
        __builtin_nontemporal_store(r5a, orow_v + 768);
        __builtin_nontemporal_store(r5b, orow_v + 1024);
    }
}

extern "C" void kernel_launch(void* const* d_in, const int* in_sizes, int n_in,
                              void* d_out, int out_size, void* d_ws, size_t ws_size,
                              hipStream_t stream) {
    const float* c   = (const float*)d_in[0];
    const float* fm3 = (const float*)d_in[1];
    const float* fm4 = (const float*)d_in[2];
    const float* fm5 = (const float*)d_in[3];
    float* out = (float*)d_out;

    dim3 grid(B_ * (V_ / VT));   // 2048 blocks
    dim3 block(NTH);             // 8 × wave32
    bilinear_concat_kernel<<<grid, block, 0, stream>>>(c, fm3, fm4, fm5, out);
}